// SAGEGRU_57226144252475
// MI455X (gfx1250) — compile-verified
//
#include <hip/hip_runtime.h>

#define N_NODES 10000
#define HGDIM   64
#define HTDIM   128
#define BATCH   8
#define TSTEPS  32
#define BT      256
#define NEDGE   160000
#define LN_EPSF 1e-5f

typedef __attribute__((ext_vector_type(2))) float v2f;
typedef __attribute__((ext_vector_type(8))) float v8f;

// ---------------- degree (once; edges fixed across all 256 slices) ------------
__global__ void deg_zero(float* deg) {
    int n = blockIdx.x * blockDim.x + threadIdx.x;
    if (n < N_NODES) deg[n] = 0.f;
}
__global__ void deg_count(const long long* __restrict__ tgt, float* deg) {
    int e = blockIdx.x * blockDim.x + threadIdx.x;
    if (e < NEDGE) atomicAdd(&deg[(int)tgt[e]], 1.0f);
}
__global__ void deg_inv(float* deg) {
    int n = blockIdx.x * blockDim.x + threadIdx.x;
    if (n < N_NODES) deg[n] = 1.0f / fmaxf(deg[n], 1.0f);
}

// ---------------- scatter-add of neighbor features (L2-resident per chunk) ----
template <int F>
__global__ void scatter_add(const float* __restrict__ xin,
                            const long long* __restrict__ src,
                            const long long* __restrict__ tgt,
                            float* __restrict__ agg) {
    constexpr int FG = F / 4;
    long long idx = (long long)blockIdx.x * blockDim.x + threadIdx.x;
    if (idx >= (long long)NEDGE * FG) return;
    int e  = (int)(idx / FG);
    int fg = (int)(idx % FG);
    int s  = blockIdx.y;  // slice within chunk
    const float* xs = xin + (size_t)s * N_NODES * F;
    float*       as = agg + (size_t)s * N_NODES * F;
    int sn = (int)src[e], tn = (int)tgt[e];
    float4 v = *(const float4*)(xs + (size_t)sn * F + fg * 4);
    float* dst = as + (size_t)tn * F + fg * 4;
    atomicAdd(dst + 0, v.x);
    atomicAdd(dst + 1, v.y);
    atomicAdd(dst + 2, v.z);
    atomicAdd(dst + 3, v.w);
}

// ---------------- fused SAGE linear (WMMA f32 16x16x4) + LayerNorm + ReLU -----
// One wave = 16 nodes x 64 outputs. 5 waves/block * 125 blocks = 625 tiles = N.
// B operands are preloaded per o-tile and the Wl/Wr products use independent
// accumulators so the WMMA chain issues back-to-back instead of waiting on a
// load per step.
template <int F>
__global__ __launch_bounds__(160) void sage_gemm_ln(
    const float* __restrict__ agg, const float* __restrict__ xin,
    const float* __restrict__ inv_deg,
    const float* __restrict__ Wl, const float* __restrict__ bl,
    const float* __restrict__ Wr, const float* __restrict__ g,
    const float* __restrict__ beta, float* __restrict__ hout) {
    constexpr int KT = F / 4;
    constexpr int RS = HGDIM + 2;  // padded row stride (bank-conflict-free)
    __shared__ float tile[5][16 * RS];

    const int lane = threadIdx.x & 31;
    const int wave = threadIdx.x >> 5;
    const int m    = lane & 15;   // node row within tile / o within o-tile
    const int hi   = lane >> 4;   // half-wave selects K pair (ISA A layout)
    const int s    = blockIdx.y;
    const int node0 = (blockIdx.x * 5 + wave) * 16;

    const float* aggRow = agg + ((size_t)s * N_NODES + node0 + m) * F;
    const float* xRow   = xin + ((size_t)s * N_NODES + node0 + m) * F;
    const float  idg    = inv_deg[node0 + m];

    v2f aA[KT], aX[KT];
#pragma unroll
    for (int kt = 0; kt < KT; ++kt) {
        int k = kt * 4 + hi * 2;  // lanes 0-15: K=0,1 ; lanes 16-31: K=2,3
        aA[kt][0] = aggRow[k] * idg;
        aA[kt][1] = aggRow[k + 1] * idg;
        aX[kt][0] = xRow[k];
        aX[kt][1] = xRow[k + 1];
    }

#pragma unroll
    for (int ot = 0; ot < 4; ++ot) {
        const int o = ot * 16 + m;
        const float* wl = Wl + (size_t)o * F;
        const float* wr = Wr + (size_t)o * F;
        // Hoist all B operands for this o-tile ahead of the WMMA chain.
        v2f bL[KT], bR[KT];
#pragma unroll
        for (int kt = 0; kt < KT; ++kt) {
            int k = kt * 4 + hi * 2;  // B[k][n] = W[o0+n][k]
            bL[kt][0] = wl[k];
            bL[kt][1] = wl[k + 1];
            bR[kt][0] = wr[k];
            bR[kt][1] = wr[k + 1];
        }
        v8f accL = {}, accR = {};
#pragma unroll
        for (int kt = 0; kt < KT; ++kt) {
            accL = __builtin_amdgcn_wmma_f32_16x16x4_f32(
                false, aA[kt], false, bL[kt], (short)0, accL, false, false);
            accR = __builtin_amdgcn_wmma_f32_16x16x4_f32(
                false, aX[kt], false, bR[kt], (short)0, accR, false, false);
        }
        const float bias = bl[o];
#pragma unroll
        for (int r = 0; r < 8; ++r)  // D: lane<16 -> M=r ; lane>=16 -> M=r+8
            tile[wave][(r + 8 * hi) * RS + ot * 16 + m] =
                accL[r] + accR[r] + bias;
    }
    __syncthreads();

    if (lane < 16) {  // one lane per node row: mean/var over 64 features
        float mu = 0.f;
#pragma unroll 8
        for (int o = 0; o < HGDIM; ++o) mu += tile[wave][lane * RS + o];
        mu *= (1.0f / HGDIM);
        float var = 0.f;
#pragma unroll 8
        for (int o = 0; o < HGDIM; ++o) {
            float d = tile[wave][lane * RS + o] - mu;
            var += d * d;
        }
        var *= (1.0f / HGDIM);
        tile[wave][lane * RS + HGDIM]     = mu;
        tile[wave][lane * RS + HGDIM + 1] = __frsqrt_rn(var + LN_EPSF);
    }
    __syncthreads();

    const int row  = lane >> 1;
    const int col0 = (lane & 1) * 32;
    const float mu = tile[wave][row * RS + HGDIM];
    const float rs = tile[wave][row * RS + HGDIM + 1];
    float* orow = hout + ((size_t)s * N_NODES + node0 + row) * HGDIM + col0;
#pragma unroll
    for (int j = 0; j < 32; ++j) {
        int o = col0 + j;
        float v = (tile[wave][row * RS + o] - mu) * rs * g[o] + beta[o];
        orow[j] = fmaxf(v, 0.f);
    }
}

// ---------------- mean over nodes -> H[bt][o] ---------------------------------
__global__ __launch_bounds__(256) void node_mean(const float* __restrict__ h,
                                                 float* __restrict__ Hout) {
    __shared__ float part[4][HGDIM];
    const int tid = threadIdx.x;
    const int o = tid & 63, p = tid >> 6;
    const float* base = h + (size_t)blockIdx.x * N_NODES * HGDIM;
    float sum = 0.f;
    for (int n = p; n < N_NODES; n += 4) sum += base[(size_t)n * HGDIM + o];
    part[p][o] = sum;
    __syncthreads();
    if (p == 0) {
        float t = part[0][o] + part[1][o] + part[2][o] + part[3][o];
        Hout[(size_t)blockIdx.x * HGDIM + o] = t * (1.0f / N_NODES);
    }
}

// ---------------- GRU over T=32 + linear head (tiny; one persistent block) ----
__global__ __launch_bounds__(256) void gru_head(
    const float* __restrict__ Hbuf, const float* __restrict__ Wih,
    const float* __restrict__ Whh, const float* __restrict__ bih,
    const float* __restrict__ bhh, const float* __restrict__ Wh,
    const float* __restrict__ bh, float* __restrict__ y) {
    __shared__ float hS[BATCH][HTDIM];
    __shared__ float xS[BATCH][HGDIM];
    __shared__ float giS[BATCH][3 * HTDIM];
    __shared__ float ghS[BATCH][3 * HTDIM];
    const int tid = threadIdx.x;
    for (int i = tid; i < BATCH * HTDIM; i += 256) hS[i / HTDIM][i % HTDIM] = 0.f;
    __syncthreads();
    for (int t = 0; t < TSTEPS; ++t) {
        for (int i = tid; i < BATCH * HGDIM; i += 256) {
            int b = i / HGDIM, f = i % HGDIM;
            xS[b][f] = Hbuf[((size_t)b * TSTEPS + t) * HGDIM + f];
        }
        __syncthreads();
        for (int i = tid; i < BATCH * 3 * HTDIM; i += 256) {
            int b = i / (3 * HTDIM), j = i % (3 * HTDIM);
            float gi = bih[j];
            const float* wi = Wih + (size_t)j * HGDIM;
#pragma unroll 8
            for (int f = 0; f < HGDIM; ++f) gi += xS[b][f] * wi[f];
            float gh = bhh[j];
            const float* wh = Whh + (size_t)j * HTDIM;
#pragma unroll 8
            for (int f = 0; f < HTDIM; ++f) gh += hS[b][f] * wh[f];
            giS[b][j] = gi;
            ghS[b][j] = gh;
        }
        __syncthreads();
        for (int i = tid; i < BATCH * HTDIM; i += 256) {
            int b = i / HTDIM, j = i % HTDIM;
            float r = 1.f / (1.f + __expf(-(giS[b][j] + ghS[b][j])));
            float z = 1.f / (1.f + __expf(-(giS[b][j + HTDIM] + ghS[b][j + HTDIM])));
            float n = tanhf(giS[b][j + 2 * HTDIM] + r * ghS[b][j + 2 * HTDIM]);
            hS[b][j] = (1.f - z) * n + z * hS[b][j];
        }
        __syncthreads();
    }
    if (tid < BATCH) {
        float acc = bh[0];
#pragma unroll 8
        for (int f = 0; f < HTDIM; ++f) acc += hS[tid][f] * Wh[f];
        y[tid] = acc;
    }
}

// ---------------- host orchestration ------------------------------------------
extern "C" void kernel_launch(void* const* d_in, const int* in_sizes, int n_in,
                              void* d_out, int out_size, void* d_ws, size_t ws_size,
                              hipStream_t stream) {
    (void)in_sizes; (void)n_in; (void)out_size;
    const float*     x_seq = (const float*)d_in[0];
    const long long* eidx  = (const long long*)d_in[1];
    const long long* src = eidx, * tgt = eidx + NEDGE;
    const float* Wl0 = (const float*)d_in[2];
    const float* bl0 = (const float*)d_in[3];
    const float* Wr0 = (const float*)d_in[4];
    const float* g0  = (const float*)d_in[5];
    const float* be0 = (const float*)d_in[6];
    const float* Wl1 = (const float*)d_in[7];
    const float* bl1 = (const float*)d_in[8];
    const float* Wr1 = (const float*)d_in[9];
    const float* g1  = (const float*)d_in[10];
    const float* be1 = (const float*)d_in[11];
    const float* Wih = (const float*)d_in[12];
    const float* Whh = (const float*)d_in[13];
    const float* bih = (const float*)d_in[14];
    const float* bhh = (const float*)d_in[15];
    const float* Wh  = (const float*)d_in[16];
    const float* bh  = (const float*)d_in[17];

    // Pick the largest chunk (slices processed concurrently) that fits d_ws;
    // keep it small enough that agg/h0/h1 for a chunk stay L2-resident.
    int ch = 32;
    while (ch > 1) {
        size_t need = (32768 + 3 * (size_t)ch * N_NODES * HGDIM) * sizeof(float);
        if (need <= ws_size) break;
        ch >>= 1;
    }
    const size_t per = (size_t)ch * N_NODES * HGDIM;
    float* inv_deg = (float*)d_ws;        // 10000 floats (padded to 16384)
    float* agg  = inv_deg + 16384;
    float* h0   = agg + per;
    float* h1   = h0 + per;
    float* Hbuf = h1 + per;               // BT*HG = 16384 floats

    deg_zero <<<(N_NODES + 255) / 256, 256, 0, stream>>>(inv_deg);
    deg_count<<<(NEDGE   + 255) / 256, 256, 0, stream>>>(tgt, inv_deg);
    deg_inv  <<<(N_NODES + 255) / 256, 256, 0, stream>>>(inv_deg);

    const int nchunk = BT / ch;
    for (int c = 0; c < nchunk; ++c) {
        const float* x0 = x_seq + (size_t)c * ch * N_NODES * 16;
        // --- layer 0 (F=16 -> 64) ---
        hipMemsetAsync(agg, 0, (size_t)ch * N_NODES * 16 * sizeof(float), stream);
        dim3 sg0((unsigned)(((long long)NEDGE * 4 + 255) / 256), ch);
        scatter_add<16><<<sg0, 256, 0, stream>>>(x0, src, tgt, agg);
        dim3 gg(125, ch);
        sage_gemm_ln<16><<<gg, 160, 0, stream>>>(agg, x0, inv_deg,
                                                 Wl0, bl0, Wr0, g0, be0, h0);
        // --- layer 1 (F=64 -> 64) ---
        hipMemsetAsync(agg, 0, (size_t)ch * N_NODES * 64 * sizeof(float), stream);
        dim3 sg1((unsigned)(((long long)NEDGE * 16 + 255) / 256), ch);
        scatter_add<64><<<sg1, 256, 0, stream>>>(h0, src, tgt, agg);
        sage_gemm_ln<64><<<gg, 160, 0, stream>>>(agg, h0, inv_deg,
                                                 Wl1, bl1, Wr1, g1, be1, h1);
        // --- mean over nodes ---
        node_mean<<<ch, 256, 0, stream>>>(h1, Hbuf + (size_t)c * ch * HGDIM);
    }
    gru_head<<<1, 256, 0, stream>>>(Hbuf, Wih, Whh, bih, bhh, Wh, bh, (float*)d_out);
}